// GraphAttentionLayer_82523501625848
// MI455X (gfx1250) — compile-verified
//
#include <hip/hip_runtime.h>

typedef __attribute__((ext_vector_type(2))) float v2f;
typedef __attribute__((ext_vector_type(8))) float v8f;

#define BATCH 8
#define NNODE 2048
#define DFEAT 64
#define NROW  (BATCH * NNODE)   // 16384 total rows
#define NWORD (NNODE / 32)      // 64 mask words per row
#define APITCH 130              // padded LDS pitch for alpha rows (8B aligned)

__device__ __forceinline__ v8f wmma_f32(v2f a, v2f b, v8f c) {
  // V_WMMA_F32_16X16X4_F32 : D(16x16 f32) = A(16x4 f32) x B(4x16 f32) + C
  return __builtin_amdgcn_wmma_f32_16x16x4_f32(false, a, false, b, (short)0, c,
                                               false, false);
}

__device__ __forceinline__ float lrelu(float z) {
  return fmaxf(z, 0.2f * z);    // == z>0 ? z : 0.2z (slope < 1)
}

// ---------------------------------------------------------------------------
// K1: hT[b, d, n] = (x[b, n, :] @ W)[d]   (16-row tile per wave, transposed
// store so K3's B operand and K1b's reduction are contiguous in n)
// ---------------------------------------------------------------------------
__global__ __launch_bounds__(32) void gat_k1_h(const float* __restrict__ x,
                                               const float* __restrict__ Wm,
                                               float* __restrict__ hT) {
  const int lane = threadIdx.x;
  const int half = lane >> 4, lr = lane & 15;
  const int tile = blockIdx.x;                       // 0..NROW/16-1 (flat b*n)
  const float* xrow = x + ((size_t)tile * 16 + lr) * DFEAT;

  v8f c0 = {}, c1 = {}, c2 = {}, c3 = {};
  for (int k0 = 0; k0 < DFEAT; k0 += 4) {
    const int ka = k0 + half * 2;                    // K pair for this half-wave
    v2f a; a.x = xrow[ka]; a.y = xrow[ka + 1];
    const float* w0 = Wm + (size_t)ka * DFEAT;       // W row ka
    const float* w1 = w0 + DFEAT;                    // W row ka+1
    v2f b0, b1, b2, b3;
    b0.x = w0[lr];      b0.y = w1[lr];
    b1.x = w0[lr + 16]; b1.y = w1[lr + 16];
    b2.x = w0[lr + 32]; b2.y = w1[lr + 32];
    b3.x = w0[lr + 48]; b3.y = w1[lr + 48];
    c0 = wmma_f32(a, b0, c0);
    c1 = wmma_f32(a, b1, c1);
    c2 = wmma_f32(a, b2, c2);
    c3 = wmma_f32(a, b3, c3);
  }
  const int bb  = tile >> 7;                         // batch
  const int nl0 = (tile & 127) * 16;                 // node base within batch
  float* tb = hT + (size_t)bb * DFEAT * NNODE;
  for (int r = 0; r < 8; ++r) {
    const int n = nl0 + r + half * 8;                // C layout: upper half M+8
    tb[(size_t)(lr)      * NNODE + n] = c0[r];
    tb[(size_t)(lr + 16) * NNODE + n] = c1[r];
    tb[(size_t)(lr + 32) * NNODE + n] = c2[r];
    tb[(size_t)(lr + 48) * NNODE + n] = c3[r];
  }
}

// ---------------------------------------------------------------------------
// K1b: e_src = h . a[:64], e_dst = h . a[64:]  (coalesced reads from hT)
// ---------------------------------------------------------------------------
__global__ void gat_k1b_proj(const float* __restrict__ hT,
                             const float* __restrict__ av,
                             float* __restrict__ es, float* __restrict__ ed) {
  const int idx = blockIdx.x * blockDim.x + threadIdx.x;   // 0..NROW-1
  const int bb = idx >> 11, n = idx & (NNODE - 1);
  const float* p = hT + (size_t)bb * DFEAT * NNODE + n;
  float s0 = 0.f, s1 = 0.f;
  for (int d = 0; d < DFEAT; ++d) {
    const float v = p[(size_t)d * NNODE];
    s0 += v * av[d];
    s1 += v * av[DFEAT + d];
  }
  es[idx] = s0;
  ed[idx] = s1;
}

// ---------------------------------------------------------------------------
// K2: per-row softmax stats (max, 1/sum) + 1-bit adjacency compression.
// Pass 1 uses monotonicity: rowmax = clamp(lrelu(e_src + max_unmasked e_dst)).
// adj (128 MB) is read from HBM exactly once; pass 2 re-reads from L0.
// ---------------------------------------------------------------------------
__global__ __launch_bounds__(256) void gat_k2_stats(
    const int* __restrict__ adj, const float* __restrict__ es,
    const float* __restrict__ ed, float* __restrict__ rmax,
    float* __restrict__ rinv, unsigned* __restrict__ maskw) {
  const int lane = threadIdx.x & 31;
  const int idx  = blockIdx.x * 8 + (threadIdx.x >> 5);  // row id, 1 wave/row
  const int bb   = idx >> 11;
  const float esv = es[idx];
  const float4* edb4  = (const float4*)(ed + (size_t)bb * NNODE);
  const int4*   arow4 = (const int4*)(adj + (size_t)idx * NNODE);
  unsigned* mrow = maskw + (size_t)idx * NWORD;

  // ---- pass 1: masked max of e_dst + mask bit emission -------------------
  float emax = -3.0e38f;
  for (int i = 0; i < NNODE / 128; ++i) {
    const int4   a4 = arow4[i * 32 + lane];            // m = i*128 + lane*4 + j
    const float4 e4 = edb4[i * 32 + lane];
    emax = fmaxf(emax, a4.x ? e4.x : -3.0e38f);
    emax = fmaxf(emax, a4.y ? e4.y : -3.0e38f);
    emax = fmaxf(emax, a4.z ? e4.z : -3.0e38f);
    emax = fmaxf(emax, a4.w ? e4.w : -3.0e38f);
    unsigned nib = (a4.x ? 1u : 0u) | (a4.y ? 2u : 0u) |
                   (a4.z ? 4u : 0u) | (a4.w ? 8u : 0u);
    unsigned word = nib << ((lane & 7) * 4);
    word |= __shfl_xor(word, 1, 32);
    word |= __shfl_xor(word, 2, 32);
    word |= __shfl_xor(word, 4, 32);
    if ((lane & 7) == 0) mrow[i * 4 + (lane >> 3)] = word;
  }
  for (int off = 16; off > 0; off >>= 1)
    emax = fmaxf(emax, __shfl_xor(emax, off, 32));
  // lrelu is monotone => rowmax over unmasked = lrelu(esv + emax);
  // clamp to -1e9 reproduces the all-masked corner exactly.
  const float mx = fmaxf(lrelu(esv + emax), -1.0e9f);

  // ---- pass 2: sum of exp(e - mx) (masked terms underflow to exactly 0) --
  float sm = 0.f;
  for (int i = 0; i < NNODE / 128; ++i) {
    const int4   a4 = arow4[i * 32 + lane];
    const float4 e4 = edb4[i * 32 + lane];
    float t;
    t = lrelu(esv + e4.x); t = a4.x ? t : -1.0e9f; sm += __expf(t - mx);
    t = lrelu(esv + e4.y); t = a4.y ? t : -1.0e9f; sm += __expf(t - mx);
    t = lrelu(esv + e4.z); t = a4.z ? t : -1.0e9f; sm += __expf(t - mx);
    t = lrelu(esv + e4.w); t = a4.w ? t : -1.0e9f; sm += __expf(t - mx);
  }
  for (int off = 16; off > 0; off >>= 1) sm += __shfl_xor(sm, off, 32);
  if (lane == 0) { rmax[idx] = mx; rinv[idx] = 1.0f / sm; }
}

// ---------------------------------------------------------------------------
// K3: h' = alpha @ h via WMMA. Block = 32 rows (2 row-tiles) x 64 cols;
// alpha chunk (32 x 128) staged in LDS from the bitmask. Per kk-step:
// 2x ds_load_b64 (A) + 1x global_load_b64 (B from hT) + 2x WMMA.
// ---------------------------------------------------------------------------
__global__ __launch_bounds__(128) void gat_k3_out(
    const float* __restrict__ hT, const unsigned* __restrict__ maskw,
    const float* __restrict__ es, const float* __restrict__ ed,
    const float* __restrict__ rmax, const float* __restrict__ rinv,
    float* __restrict__ out) {
  __shared__ float s_alpha[32 * APITCH];            // [row][m], pitch 130
  __shared__ float s_es[32], s_mx[32], s_inv[32];

  const int tid  = threadIdx.x;
  const int tile = blockIdx.x;                      // 0..NROW/32-1
  const int bb   = tile >> 6;                       // / (NNODE/32)
  const int rowbase = tile * 32;                    // flat row base
  if (tid < 32) {
    s_es[tid]  = es[rowbase + tid];
    s_mx[tid]  = rmax[rowbase + tid];
    s_inv[tid] = rinv[rowbase + tid];
  }
  const int wave = tid >> 5, lane = tid & 31, half = lane >> 4, lr = lane & 15;
  const float* edb = ed + (size_t)bb * NNODE;
  const float* hTb = hT + (size_t)bb * DFEAT * NNODE;
  const unsigned* mbase = maskw + (size_t)rowbase * NWORD;
  const float* bcol = hTb + (size_t)(wave * 16 + lr) * NNODE;  // B column base

  v8f c0 = {}, c1 = {};
  for (int chunk = 0; chunk < NNODE; chunk += 128) {
    __syncthreads();                                // consts ready / LDS reusable
    const int m = chunk + tid;
    const float edv = edb[m];
    const int widx = m >> 5;
    const unsigned bit = 1u << (m & 31);
    for (int r = 0; r < 32; ++r) {
      float e = lrelu(s_es[r] + edv);
      e = (mbase[(size_t)r * NWORD + widx] & bit) ? e : -1.0e9f;
      s_alpha[r * APITCH + tid] = __expf(e - s_mx[r]) * s_inv[r];
    }
    __syncthreads();
    for (int kk = 0; kk < 128; kk += 4) {
      const int ml = kk + half * 2;                 // K pair for this half-wave
      const v2f a0 = *(const v2f*)&s_alpha[lr * APITCH + ml];        // rows 0-15
      const v2f a1 = *(const v2f*)&s_alpha[(16 + lr) * APITCH + ml]; // rows 16-31
      const v2f bv = *(const v2f*)&bcol[chunk + ml];                 // b64 load
      c0 = wmma_f32(a0, bv, c0);
      c1 = wmma_f32(a1, bv, c1);
    }
  }
  float* ob = out + (size_t)rowbase * DFEAT + wave * 16 + lr;
  for (int r = 0; r < 8; ++r) {
    ob[(size_t)(r + half * 8) * DFEAT]      = c0[r];
    ob[(size_t)(16 + r + half * 8) * DFEAT] = c1[r];
  }
}

// ---------------------------------------------------------------------------
extern "C" void kernel_launch(void* const* d_in, const int* in_sizes, int n_in,
                              void* d_out, int out_size, void* d_ws,
                              size_t ws_size, hipStream_t stream) {
  (void)in_sizes; (void)n_in; (void)out_size; (void)ws_size;
  const float* x   = (const float*)d_in[0];   // (8,2048,64) f32
  const int*   adj = (const int*)d_in[1];     // (8,2048,2048) i32
  const float* Wm  = (const float*)d_in[2];   // (64,64) f32
  const float* av  = (const float*)d_in[3];   // (128,) f32
  float* out = (float*)d_out;                 // (8,2048,64) f32

  float* ws   = (float*)d_ws;                 // ~8.3 MB used
  float* hT   = ws;                           // NROW*DFEAT floats (4 MB), [b][d][n]
  float* es   = hT + (size_t)NROW * DFEAT;    // NROW
  float* ed   = es + NROW;                    // NROW
  float* rmx  = ed + NROW;                    // NROW
  float* rinv = rmx + NROW;                   // NROW
  unsigned* maskw = (unsigned*)(rinv + NROW); // NROW*NWORD words (4 MB)

  hipLaunchKernelGGL(gat_k1_h, dim3(NROW / 16), dim3(32), 0, stream, x, Wm, hT);
  hipLaunchKernelGGL(gat_k1b_proj, dim3(NROW / 256), dim3(256), 0, stream, hT,
                     av, es, ed);
  hipLaunchKernelGGL(gat_k2_stats, dim3(NROW / 8), dim3(256), 0, stream, adj, es,
                     ed, rmx, rinv, maskw);
  hipLaunchKernelGGL(gat_k3_out, dim3(NROW / 32), dim3(128), 0, stream, hT,
                     maskw, es, ed, rmx, rinv, out);
}